// TensorProduct_78056735638120
// MI455X (gfx1250) — compile-verified
//
#include <hip/hip_runtime.h>

// Memory-bound "uvu" tensor product for MI455X (gfx1250, wave32).
// One wave32 = one batch row; lane = channel u (MUL=32).
// ~7.1 KB HBM traffic per row -> ~928 MB total -> ~40us floor @ 23.3 TB/s.
//
// CDNA5 data-movement: each wave double-buffers its next row's x (1152B) and
// W (1408B) into LDS with GLOBAL_LOAD_ASYNC_TO_LDS_B128 (ASYNCcnt pipeline,
// s_wait_asynccnt), computes the current row from LDS (ds_load_b96/b128,
// bank-conflict-free: strides 3,5 coprime with 64 banks), and streams results
// out with non-temporal stores so the single-touch data never thrashes L2.

#define BLOCK 256
#define WAVES_PER_BLOCK 8

#if defined(__AMDGCN__) && __has_builtin(__builtin_amdgcn_global_load_async_to_lds_b128)
#define USE_ASYNC 1
#else
#define USE_ASYNC 0
#endif

// The builtin expects pointers to 16-byte int vectors:
//   arg0: v4i in address space 1 (global), arg1: v4i in address space 3 (LDS)
typedef int v4i_t __attribute__((vector_size(16)));
typedef __attribute__((address_space(1))) v4i_t gv4i_t;
typedef __attribute__((address_space(3))) v4i_t lv4i_t;

#if defined(__AMDGCN__) && __has_builtin(__builtin_amdgcn_s_wait_asynccnt)
#define WAIT_ASYNC(n) __builtin_amdgcn_s_wait_asynccnt(n)
#elif defined(__AMDGCN__)
#define WAIT_ASYNC(n) asm volatile("s_wait_asynccnt %0" ::"n"(n))
#else
#define WAIT_ASYNC(n)
#endif

__device__ __forceinline__ void nts(float* p, float v) {
  __builtin_nontemporal_store(v, p);
}

// Per-row compute: xb = 288-float row (global or LDS), wrow = 352-float row,
// yb = 9-float row (global, wave-uniform -> scalar loads), ob = 1120-float out.
__device__ __forceinline__ void tp_row(const float* __restrict__ xb,
                                       const float* __restrict__ wrow,
                                       const float* __restrict__ yb,
                                       float* __restrict__ ob, int lane) {
  // Real-basis Wigner-3j constants (reference convention, signs included)
  constexpr float S3  = 0.57735026918962576f;  // 1/sqrt(3)
  constexpr float S5  = 0.44721359549995794f;  // 1/sqrt(5)
  constexpr float S10 = 0.31622776601683794f;  // 1/sqrt(10)
  constexpr float S30 = 0.18257418583505536f;  // 1/sqrt(30)
  constexpr float A70 = 0.23904572186687872f;  // 2/sqrt(70)
  constexpr float B70 = 0.11952286093343936f;  // 1/sqrt(70)
  constexpr float T70 = 0.20701966780270626f;  // sqrt(3/70)

  const float x0 = xb[lane];
  const float* x1p = xb + 32 + 3 * lane;
  const float x10 = x1p[0], x11 = x1p[1], x12 = x1p[2];
  const float* x2p = xb + 128 + 5 * lane;
  const float x20 = x2p[0], x21 = x2p[1], x22 = x2p[2], x23 = x2p[3],
              x24 = x2p[4];

  const float y0  = yb[0];
  const float y10 = yb[1], y11 = yb[2], y12 = yb[3];
  const float y20 = yb[4], y21 = yb[5], y22 = yb[6], y23 = yb[7], y24 = yb[8];

  const float w0  = wrow[0 * 32 + lane];
  const float w1  = wrow[1 * 32 + lane];
  const float w2  = wrow[2 * 32 + lane];
  const float w3  = wrow[3 * 32 + lane];
  const float w4  = wrow[4 * 32 + lane];
  const float w5  = wrow[5 * 32 + lane];
  const float w6  = wrow[6 * 32 + lane];
  const float w7  = wrow[7 * 32 + lane];
  const float w8  = wrow[8 * 32 + lane];
  const float w9  = wrow[9 * 32 + lane];
  const float w10 = wrow[10 * 32 + lane];

  // path 0: (0,0,0)
  nts(ob + lane, w0 * x0 * y0);
  // path 1: (0,1,1) = -1/sqrt(3) * I
  {
    const float f = -S3 * (w1 * x0);
    float* o = ob + 32 + 3 * lane;
    nts(o + 0, f * y10); nts(o + 1, f * y11); nts(o + 2, f * y12);
  }
  // path 2: (0,2,2) = +1/sqrt(5) * I
  {
    const float f = S5 * (w2 * x0);
    float* o = ob + 128 + 5 * lane;
    nts(o + 0, f * y20); nts(o + 1, f * y21); nts(o + 2, f * y22);
    nts(o + 3, f * y23); nts(o + 4, f * y24);
  }
  // path 3: (1,0,1) = -1/sqrt(3) * I
  {
    const float f = -S3 * (w3 * y0);
    float* o = ob + 288 + 3 * lane;
    nts(o + 0, f * x10); nts(o + 1, f * x11); nts(o + 2, f * x12);
  }
  // path 4: (1,1,0) = -1/sqrt(3) * dot
  nts(ob + 384 + lane, -S3 * w4 * (x10 * y10 + x11 * y11 + x12 * y12));
  // path 5: (1,1,2)
  {
    float* o = ob + 416 + 5 * lane;
    nts(o + 0, w5 * (S10 * (x10 * y12 + x12 * y10)));
    nts(o + 1, w5 * (S10 * (x10 * y11 + x11 * y10)));
    nts(o + 2, w5 * (S30 * (2.0f * x11 * y11 - x10 * y10 - x12 * y12)));
    nts(o + 3, w5 * (S10 * (x11 * y12 + x12 * y11)));
    nts(o + 4, w5 * (S10 * (x12 * y12 - x10 * y10)));
  }
  // path 6: (1,2,1)
  {
    float* o = ob + 576 + 3 * lane;
    nts(o + 0, w6 * (S10 * (x12 * y20 + x11 * y21 - x10 * y24) - S30 * (x10 * y22)));
    nts(o + 1, w6 * (S10 * (x10 * y21 + x12 * y23) + 2.0f * S30 * (x11 * y22)));
    nts(o + 2, w6 * (S10 * (x10 * y20 + x11 * y23 + x12 * y24) - S30 * (x12 * y22)));
  }
  // path 7: (2,0,2) = +1/sqrt(5) * I
  {
    const float f = S5 * (w7 * y0);
    float* o = ob + 672 + 5 * lane;
    nts(o + 0, f * x20); nts(o + 1, f * x21); nts(o + 2, f * x22);
    nts(o + 3, f * x23); nts(o + 4, f * x24);
  }
  // path 8: (2,1,1)
  {
    float* o = ob + 832 + 3 * lane;
    nts(o + 0, w8 * (S10 * (x21 * y11 - x24 * y10) - S30 * (x22 * y10)));
    nts(o + 1, w8 * (S10 * (x21 * y10 + x23 * y12) + 2.0f * S30 * (x22 * y11)));
    nts(o + 2, w8 * (S10 * (x20 * y10 + x23 * y11 + x24 * y12) - S30 * (x22 * y12)));
  }
  // path 9: (2,2,0) = +1/sqrt(5) * dot
  nts(ob + 928 + lane,
      S5 * w9 * (x20 * y20 + x21 * y21 + x22 * y22 + x23 * y23 + x24 * y24));
  // path 10: (2,2,2) fully symmetric real w3j
  {
    float* o = ob + 960 + 5 * lane;
    nts(o + 0, w10 * (A70 * (x20 * y22 + x22 * y20)
                    - T70 * (x21 * y23 + x23 * y21)));
    nts(o + 1, w10 * (T70 * (x21 * y24 + x24 * y21 - x20 * y23 - x23 * y20)
                    - B70 * (x21 * y22 + x22 * y21)));
    nts(o + 2, w10 * (A70 * (x20 * y20 - x22 * y22 + x24 * y24)
                    - B70 * (x21 * y21 + x23 * y23)));
    nts(o + 3, w10 * (-B70 * (x22 * y23 + x23 * y22)
                    -  T70 * (x23 * y24 + x24 * y23 + x20 * y21 + x21 * y20)));
    nts(o + 4, w10 * (A70 * (x22 * y24 + x24 * y22)
                    + T70 * (x21 * y21 - x23 * y23)));
  }
}

#if USE_ASYNC
__device__ __forceinline__ void async_b128(const float* g, float* l) {
  __builtin_amdgcn_global_load_async_to_lds_b128((gv4i_t*)g, (lv4i_t*)l, 0, 0);
}

// Stage one row (x: 288 floats = 1152B, W: 352 floats = 1408B) into LDS with
// six fire-and-forget B128 async DMA ops (ASYNCcnt += 6). 16B-aligned:
// row strides 1152B/1408B are multiples of 16.
__device__ __forceinline__ void issue_row(const float* xg, const float* wg,
                                          float* lx, float* lw, int lane) {
  const int o = lane * 4;  // floats; 16B per lane
  async_b128(xg + o, lx + o);
  async_b128(xg + 128 + o, lx + 128 + o);
  if (lane < 8)  // bytes 1024..1151
    async_b128(xg + 256 + o, lx + 256 + o);
  async_b128(wg + o, lw + o);
  async_b128(wg + 128 + o, lw + 128 + o);
  if (lane < 24)  // bytes 1024..1407
    async_b128(wg + 256 + o, lw + 256 + o);
}
#endif

__global__ __launch_bounds__(BLOCK) void tp_uvu_kernel(
    const float* __restrict__ x,   // [nrows, 288]
    const float* __restrict__ y,   // [nrows, 9]
    const float* __restrict__ W,   // [nrows, 352]
    float* __restrict__ out,       // [nrows, 1120]
    int nrows) {
  const int lane    = (int)(threadIdx.x & 31);
  const int wid     = __builtin_amdgcn_readfirstlane((int)(threadIdx.x >> 5));
  const int wgid    = (int)blockIdx.x * WAVES_PER_BLOCK + wid;
  const int wstride = (int)gridDim.x * WAVES_PER_BLOCK;

#if USE_ASYNC
  // Per-wave double buffer: [x 288 | W 352] x 2 = 5120B; 8 waves -> 40KB LDS.
  __shared__ __align__(16) float smem[WAVES_PER_BLOCK * 2 * 640];
  float* mybuf = smem + wid * (2 * 640);

  int cur = 0;
  if (wgid < nrows)
    issue_row(x + (size_t)wgid * 288, W + (size_t)wgid * 352,
              mybuf, mybuf + 288, lane);

  for (int b = wgid; b < nrows; b += wstride) {
    const int bn = b + wstride;
    float* bc = mybuf + cur * 640;
    float* bx = mybuf + (cur ^ 1) * 640;
    if (bn < nrows) {
      // Issue next row's DMA before consuming current: ASYNCcnt rises to 12,
      // wait <=6 releases exactly the current row's six loads (in-order).
      issue_row(x + (size_t)bn * 288, W + (size_t)bn * 352, bx, bx + 288, lane);
      WAIT_ASYNC(6);
    } else {
      WAIT_ASYNC(0);
    }
    asm volatile("" ::: "memory");  // compiler fence: LDS now holds row b
    tp_row(bc, bc + 288, y + (size_t)b * 9, out + (size_t)b * 1120, lane);
    cur ^= 1;
  }
#else
  // Fallback: direct coalesced global loads + explicit prefetch of next row.
  for (int b = wgid; b < nrows; b += wstride) {
    const int bn = b + wstride;
    if (bn < nrows) {
      __builtin_prefetch(x + (size_t)bn * 288 + lane * 9, 0, 1);
      __builtin_prefetch(W + (size_t)bn * 352 + lane * 11, 0, 1);
      __builtin_prefetch(y + (size_t)bn * 9, 0, 1);
    }
    tp_row(x + (size_t)b * 288, W + (size_t)b * 352,
           y + (size_t)b * 9, out + (size_t)b * 1120, lane);
  }
#endif
}

extern "C" void kernel_launch(void* const* d_in, const int* in_sizes, int n_in,
                              void* d_out, int out_size, void* d_ws, size_t ws_size,
                              hipStream_t stream) {
  const float* x = (const float*)d_in[0];
  const float* y = (const float*)d_in[1];
  const float* W = (const float*)d_in[2];
  float* out = (float*)d_out;

  const int nrows = in_sizes[0] / 288;  // B (=131072 in reference)

  // ~4 rows per wave so the ASYNCcnt double-buffer pipeline has depth.
  int waves = (nrows + 3) / 4;
  int blocks = (waves + WAVES_PER_BLOCK - 1) / WAVES_PER_BLOCK;
  if (blocks < 1) blocks = 1;
  if (blocks > 16384) blocks = 16384;

  tp_uvu_kernel<<<blocks, BLOCK, 0, stream>>>(x, y, W, out, nrows);
}